// SE3TransformerWrapper_37417755083160
// MI455X (gfx1250) — compile-verified
//
#include <hip/hip_runtime.h>
#include <hip/hip_bf16.h>
#include <math.h>

#define NNODES 16384
#define NEDGES 393216

typedef __attribute__((ext_vector_type(2))) float v2f;
typedef __attribute__((ext_vector_type(8))) float v8f;

// ---------------------------------------------------------------------------
// Monotonic float <-> uint encoding for atomicMax-based segment max
// ---------------------------------------------------------------------------
__device__ __forceinline__ unsigned int fkey(float f) {
  int i = __float_as_int(f);
  return (i >= 0) ? ((unsigned int)i | 0x80000000u) : ~(unsigned int)i;
}
__device__ __forceinline__ float fdec(unsigned int k) {
  int i = (k & 0x80000000u) ? (int)(k & 0x7fffffffu) : (int)(~k);
  return __int_as_float(i);
}

// ---------------------------------------------------------------------------
// Fused node GEMMs (fast path): D[w] = A @ B[w], A: M x 32 (row-major,
// contiguous), B[w]: 32 x 32, D[w]: M x 32.  One wave per 16x16 tile
// position; A fragments loaded ONCE (aligned b64 loads) and reused across
// all weights -> 8 v_wmma_f32_16x16x4_f32 per weight, branch-free.
// ---------------------------------------------------------------------------
struct GemmPtrs {
  const float* B[5];
  float* D[5];
};

__global__ void node_gemm_fused(const float* __restrict__ A, GemmPtrs p,
                                int nW, int M) {
  int waveId = blockIdx.x * blockDim.y + threadIdx.y;
  int tm = waveId >> 1;   // tilesN == 2 (32 output columns)
  int tn = waveId & 1;
  if (tm * 16 >= M) return;  // wave-uniform
  int lane = threadIdx.x;
  int half = lane >> 4;
  int l16 = lane & 15;

  // A fragments for K=0..31: lane holds A[row][k0+2*half], A[row][k0+2*half+1]
  const float* arow = A + (long)(tm * 16 + l16) * 32 + 2 * half;
  v2f a[8];
#pragma unroll
  for (int k = 0; k < 8; ++k) a[k] = *(const v2f*)(arow + k * 4);

  int col = tn * 16 + l16;
  for (int w = 0; w < nW; ++w) {  // uniform loop
    const float* Bp = p.B[w];
    v8f acc = {};
#pragma unroll
    for (int k = 0; k < 8; ++k) {
      int kk = k * 4 + 2 * half;
      v2f b;
      b.x = Bp[kk * 32 + col];
      b.y = Bp[(kk + 1) * 32 + col];
      acc = __builtin_amdgcn_wmma_f32_16x16x4_f32(
          false, a[k], false, b, (short)0, acc, false, false);
    }
    float* Dp = p.D[w];
#pragma unroll
    for (int v = 0; v < 8; ++v)
      Dp[(long)(tm * 16 + v + 8 * half) * 32 + col] = acc[v];
  }
}

// ---------------------------------------------------------------------------
// Generic fp32 WMMA GEMM (strided, masked K/N) for the einsum slices and
// narrow outputs.  Masking done with clamped unconditional loads + selects
// (no divergent exec branches in the k-loop).
// ---------------------------------------------------------------------------
__global__ void wmma_gemm_f32(const float* __restrict__ A, int lda, int acs,
                              const float* __restrict__ B, int ldb, int bcs,
                              float* __restrict__ D, int ldd, int dcs,
                              int M, int K, int Nc, int tilesN) {
  int waveId = blockIdx.x * blockDim.y + threadIdx.y;
  int tm = waveId / tilesN;
  int tn = waveId % tilesN;
  if (tm * 16 >= M) return;  // wave-uniform
  int lane = threadIdx.x;
  int half = lane >> 4;
  int l16 = lane & 15;
  long arow = (long)(tm * 16 + l16);
  int bcol = tn * 16 + l16;
  int bcolc = min(bcol, Nc - 1);
  bool colOK = (bcol < Nc);
  v8f acc = {};
  int kEnd = (K + 3) & ~3;
  for (int k0 = 0; k0 < kEnd; k0 += 4) {
    int ka = k0 + 2 * half;
    int ka0 = min(ka, K - 1);
    int ka1 = min(ka + 1, K - 1);
    v2f a, b;
    float ax = A[arow * lda + (long)ka0 * acs];
    float ay = A[arow * lda + (long)ka1 * acs];
    float bx = B[(long)ka0 * ldb + (long)bcolc * bcs];
    float by = B[(long)ka1 * ldb + (long)bcolc * bcs];
    a.x = (ka < K) ? ax : 0.0f;
    a.y = (ka + 1 < K) ? ay : 0.0f;
    b.x = (ka < K && colOK) ? bx : 0.0f;
    b.y = (ka + 1 < K && colOK) ? by : 0.0f;
    acc = __builtin_amdgcn_wmma_f32_16x16x4_f32(
        false, a, false, b, (short)0, acc, false, false);
  }
#pragma unroll
  for (int v = 0; v < 8; ++v) {
    long r = (long)(tm * 16 + v + 8 * half);
    if (colOK) D[r * ldd + (long)bcol * dcs] = acc[v];
  }
}

// ---------------------------------------------------------------------------
// Edge geometry (shared by both layers): d = |xyz[dst]-xyz[src]|, rhat
// ---------------------------------------------------------------------------
__global__ void edge_geom_kernel(const float* __restrict__ xyz,
                                 const int* __restrict__ src,
                                 const int* __restrict__ dst,
                                 float* __restrict__ dlen,
                                 float* __restrict__ rhat, int E) {
  int e = blockIdx.x * blockDim.x + threadIdx.x;
  if (e >= E) return;
  int s = src[e], t = dst[e];
  float rx = xyz[t * 3 + 0] - xyz[s * 3 + 0];
  float ry = xyz[t * 3 + 1] - xyz[s * 3 + 1];
  float rz = xyz[t * 3 + 2] - xyz[s * 3 + 2];
  float d = sqrtf(rx * rx + ry * ry + rz * rz);
  float inv = 1.0f / (d + 1e-8f);
  dlen[e] = d;
  rhat[e * 3 + 0] = rx * inv;
  rhat[e * 3 + 1] = ry * inv;
  rhat[e * 3 + 2] = rz * inv;
}

// ---------------------------------------------------------------------------
// Init segment-max / segment-sum buffers
// ---------------------------------------------------------------------------
__global__ void init_mz_kernel(unsigned int* __restrict__ m,
                               float* __restrict__ z, int n) {
  int i = blockIdx.x * blockDim.x + threadIdx.x;
  if (i >= n) return;
  m[i] = 0u;  // minimum key == most negative float
  z[i] = 0.0f;
}

// ---------------------------------------------------------------------------
// Per-edge: radial MLP g, h1p = h1[src].rhat, k0, attention logits,
// segment max via atomicMax
// ---------------------------------------------------------------------------
template <int CIN1>
__global__ void edge_logits_kernel(const float* __restrict__ ef,
                                   const float* __restrict__ dlen,
                                   const float* __restrict__ rhat,
                                   const int* __restrict__ src,
                                   const int* __restrict__ dst,
                                   const float* __restrict__ h1,
                                   const float* __restrict__ Wr1,
                                   const float* __restrict__ br1,
                                   const float* __restrict__ Wr2,
                                   const float* __restrict__ Wk10,
                                   const float* __restrict__ Qn,
                                   const float* __restrict__ K00n,
                                   float* __restrict__ gbuf,
                                   float* __restrict__ h1pbuf,
                                   float* __restrict__ logitbuf,
                                   unsigned int* __restrict__ mbuf, int E) {
  __shared__ float sWr1[33 * 32];
  __shared__ float sbr[32];
  __shared__ float sWr2[32 * 4];
  __shared__ float sWk10[CIN1 * 32];
  for (int i = threadIdx.x; i < 33 * 32; i += blockDim.x) sWr1[i] = Wr1[i];
  for (int i = threadIdx.x; i < 32; i += blockDim.x) sbr[i] = br1[i];
  for (int i = threadIdx.x; i < 32 * 4; i += blockDim.x) sWr2[i] = Wr2[i];
  for (int i = threadIdx.x; i < CIN1 * 32; i += blockDim.x) sWk10[i] = Wk10[i];
  __syncthreads();

  int e = blockIdx.x * blockDim.x + threadIdx.x;
  if (e >= E) return;
  int s = src[e], t = dst[e];
  float d = dlen[e];

  float efr[32];
#pragma unroll
  for (int j = 0; j < 32; ++j) efr[j] = ef[(long)e * 32 + j];

  // g = relu([ef, d] @ Wr1 + br1) @ Wr2
  float g[4] = {0.f, 0.f, 0.f, 0.f};
  for (int c = 0; c < 32; ++c) {
    float acc = sbr[c] + d * sWr1[32 * 32 + c];
#pragma unroll
    for (int j = 0; j < 32; ++j) acc += efr[j] * sWr1[j * 32 + c];
    acc = fmaxf(acc, 0.0f);
    g[0] += acc * sWr2[c * 4 + 0];
    g[1] += acc * sWr2[c * 4 + 1];
    g[2] += acc * sWr2[c * 4 + 2];
    g[3] += acc * sWr2[c * 4 + 3];
  }
#pragma unroll
  for (int k = 0; k < 4; ++k) gbuf[(long)e * 4 + k] = g[k];

  float rh0 = rhat[e * 3 + 0], rh1 = rhat[e * 3 + 1], rh2 = rhat[e * 3 + 2];
  float h1p[CIN1];
#pragma unroll
  for (int j = 0; j < CIN1; ++j) {
    const float* hp = h1 + ((long)s * CIN1 + j) * 3;
    float v = hp[0] * rh0 + hp[1] * rh1 + hp[2] * rh2;
    h1p[j] = v;
    h1pbuf[(long)e * 32 + j] = v;
  }

  // logits[h] = sum_c q[c]*k0[c] / sqrt(8) for c in head h
  float lg[4] = {0.f, 0.f, 0.f, 0.f};
  for (int c = 0; c < 32; ++c) {
    float kd = 0.0f;
#pragma unroll
    for (int j = 0; j < CIN1; ++j) kd += h1p[j] * sWk10[j * 32 + c];
    float kc = g[0] * K00n[(long)s * 32 + c] + g[2] * kd;
    lg[c >> 3] += Qn[(long)t * 32 + c] * kc;
  }
  const float inv = 0.35355339059327373f;  // 1/sqrt(8)
#pragma unroll
  for (int h = 0; h < 4; ++h) {
    float L = lg[h] * inv;
    logitbuf[(long)e * 4 + h] = L;
    atomicMax(&mbuf[(long)t * 4 + h], fkey(L));
  }
}

// ---------------------------------------------------------------------------
// Per-edge: a = exp(logit - m[dst]); z segment-sum (atomicAdd). a in-place.
// ---------------------------------------------------------------------------
__global__ void edge_softmax_kernel(const int* __restrict__ dst,
                                    float* att,  // logits in, a out (in place)
                                    const unsigned int* __restrict__ mbuf,
                                    float* __restrict__ zbuf, int E) {
  int e = blockIdx.x * blockDim.x + threadIdx.x;
  if (e >= E) return;
  int t = dst[e];
#pragma unroll
  for (int h = 0; h < 4; ++h) {
    float m = fdec(mbuf[(long)t * 4 + h]);
    float a = __expf(att[(long)e * 4 + h] - m);
    att[(long)e * 4 + h] = a;
    atomicAdd(&zbuf[(long)t * 4 + h], a);
  }
}

// ---------------------------------------------------------------------------
// Per-edge scatter: alpha, v0, v1 -> atomicAdd into out0 (=h0@Ws0) and
// out1 (=einsum(h1,Ws1)) accumulators.
// ---------------------------------------------------------------------------
template <int CIN1, int COUT1>
__global__ void edge_scatter_kernel(const int* __restrict__ src,
                                    const int* __restrict__ dst,
                                    const float* __restrict__ rhat,
                                    const float* __restrict__ gbuf,
                                    const float* __restrict__ h1pbuf,
                                    const float* __restrict__ abuf,
                                    const float* __restrict__ zbuf,
                                    const float* __restrict__ V00n,
                                    const float* __restrict__ V01n,
                                    const float* __restrict__ V11n,
                                    const float* __restrict__ Wv10,
                                    float* __restrict__ out0,
                                    float* __restrict__ out1, int E) {
  __shared__ float sWv10[CIN1 * 32];
  for (int i = threadIdx.x; i < CIN1 * 32; i += blockDim.x) sWv10[i] = Wv10[i];
  __syncthreads();

  int e = blockIdx.x * blockDim.x + threadIdx.x;
  if (e >= E) return;
  int s = src[e], t = dst[e];

  float alpha[4];
  float am = 0.0f;
#pragma unroll
  for (int h = 0; h < 4; ++h) {
    alpha[h] = abuf[(long)e * 4 + h] / (zbuf[(long)t * 4 + h] + 1e-9f);
    am += alpha[h];
  }
  am *= 0.25f;

  float g0 = gbuf[(long)e * 4 + 0];
  float g1 = gbuf[(long)e * 4 + 1];
  float g2 = gbuf[(long)e * 4 + 2];
  float g3 = gbuf[(long)e * 4 + 3];

  float h1p[CIN1];
#pragma unroll
  for (int j = 0; j < CIN1; ++j) h1p[j] = h1pbuf[(long)e * 32 + j];

  for (int c = 0; c < 32; ++c) {
    float vd = 0.0f;
#pragma unroll
    for (int j = 0; j < CIN1; ++j) vd += h1p[j] * sWv10[j * 32 + c];
    float v0c = g0 * V00n[(long)s * 32 + c] + g2 * vd;
    atomicAdd(&out0[(long)t * 32 + c], alpha[c >> 3] * v0c);
  }

  float rh[3] = {rhat[e * 3 + 0], rhat[e * 3 + 1], rhat[e * 3 + 2]};
#pragma unroll
  for (int dch = 0; dch < COUT1; ++dch) {
    float a01 = g1 * V01n[(long)s * COUT1 + dch];
    long bs = ((long)s * COUT1 + dch) * 3;
    long bt = ((long)t * COUT1 + dch) * 3;
#pragma unroll
    for (int i = 0; i < 3; ++i) {
      float v1 = a01 * rh[i] + g3 * V11n[bs + i];
      atomicAdd(&out1[bt + i], am * v1);
    }
  }
}

// ---------------------------------------------------------------------------
// Per-node finalize: layernorm+relu on type-0, RMS norm on type-1 (if act)
// ---------------------------------------------------------------------------
__global__ void node_finalize_kernel(const float* __restrict__ acc0,
                                     const float* __restrict__ acc1, int cout1,
                                     float* __restrict__ o0,
                                     float* __restrict__ o1, int applyAct,
                                     int Nn) {
  int n = blockIdx.x * blockDim.x + threadIdx.x;
  if (n >= Nn) return;

  float v[32];
#pragma unroll
  for (int c = 0; c < 32; ++c) v[c] = acc0[(long)n * 32 + c];
  if (applyAct) {
    float mu = 0.0f;
#pragma unroll
    for (int c = 0; c < 32; ++c) mu += v[c];
    mu *= (1.0f / 32.0f);
    float var = 0.0f;
#pragma unroll
    for (int c = 0; c < 32; ++c) {
      float dv = v[c] - mu;
      var += dv * dv;
    }
    var *= (1.0f / 32.0f);
    float inv = rsqrtf(var + 1e-5f);
#pragma unroll
    for (int c = 0; c < 32; ++c)
      o0[(long)n * 32 + c] = fmaxf((v[c] - mu) * inv, 0.0f);
  } else {
#pragma unroll
    for (int c = 0; c < 32; ++c) o0[(long)n * 32 + c] = v[c];
  }

  long base = (long)n * cout1 * 3;
  if (applyAct) {
    float ss = 0.0f;
    for (int i = 0; i < cout1 * 3; ++i) {
      float x = acc1[base + i];
      ss += x * x;
    }
    float rms = sqrtf(ss / (float)cout1 + 1e-5f);
    float inv = 1.0f / rms;
    for (int i = 0; i < cout1 * 3; ++i) o1[base + i] = acc1[base + i] * inv;
  } else {
    for (int i = 0; i < cout1 * 3; ++i) o1[base + i] = acc1[base + i];
  }
}

// ---------------------------------------------------------------------------
// Host orchestration
// ---------------------------------------------------------------------------
extern "C" void kernel_launch(void* const* d_in, const int* in_sizes, int n_in,
                              void* d_out, int out_size, void* d_ws,
                              size_t ws_size, hipStream_t stream) {
  const int N = NNODES;
  const int E = NEDGES;

  const float* xyz = (const float*)d_in[0];
  const float* h0in = (const float*)d_in[1];
  const float* h1in = (const float*)d_in[2];
  const float* ef = (const float*)d_in[3];
  const int* esrc = (const int*)d_in[n_in - 2];
  const int* edst = (const int*)d_in[n_in - 1];

  // Workspace layout (floats)
  float* ws = (float*)d_ws;
  size_t off = 0;
  auto alloc = [&](size_t n) { float* p = ws + off; off += n; return p; };
  float* dE = alloc((size_t)E);
  float* rhat = alloc((size_t)E * 3);
  float* gbuf = alloc((size_t)E * 4);
  float* h1pbuf = alloc((size_t)E * 32);
  float* attbuf = alloc((size_t)E * 4);
  float* Qn = alloc((size_t)N * 32);
  float* K00n = alloc((size_t)N * 32);
  float* V00n = alloc((size_t)N * 32);
  float* V01n = alloc((size_t)N * 32);
  float* S0 = alloc((size_t)N * 32);  // out0 accumulator
  float* V11n = alloc((size_t)N * 96);
  float* S1 = alloc((size_t)N * 96);  // out1 accumulator
  unsigned int* mbuf = (unsigned int*)alloc((size_t)N * 4);
  float* zbuf = alloc((size_t)N * 4);
  float* h0mid = alloc((size_t)N * 32);
  float* h1mid = alloc((size_t)N * 96);

  dim3 eblk(256), egrd((E + 255) / 256);
  dim3 nblk(256), ngrd((N + 255) / 256);

  edge_geom_kernel<<<egrd, eblk, 0, stream>>>(xyz, esrc, edst, dE, rhat, E);

  auto gemm = [&](const float* A, int lda, int acs, const float* B, int ldb,
                  int bcs, float* D, int ldd, int dcs, int M, int K, int Nc) {
    int tilesN = (Nc + 15) / 16;
    int tiles = (M / 16) * tilesN;
    dim3 blk(32, 4);
    dim3 grd((tiles + 3) / 4);
    wmma_gemm_f32<<<grd, blk, 0, stream>>>(A, lda, acs, B, ldb, bcs, D, ldd,
                                           dcs, M, K, Nc, tilesN);
  };

  for (int l = 0; l < 2; ++l) {
    const int P = 4 + l * 12;  // sorted-key leaf order per layer dict
    const float* Wk00 = (const float*)d_in[P + 0];
    const float* Wk10 = (const float*)d_in[P + 1];
    const float* Wq = (const float*)d_in[P + 2];
    const float* Wr1 = (const float*)d_in[P + 3];
    const float* Wr2 = (const float*)d_in[P + 4];
    const float* Ws0 = (const float*)d_in[P + 5];
    const float* Ws1 = (const float*)d_in[P + 6];
    const float* Wv00 = (const float*)d_in[P + 7];
    const float* Wv01 = (const float*)d_in[P + 8];
    const float* Wv10 = (const float*)d_in[P + 9];
    const float* Wv11 = (const float*)d_in[P + 10];
    const float* br1 = (const float*)d_in[P + 11];

    const int cin1 = (l == 0) ? 3 : 32;
    const int cout1 = (l == 0) ? 32 : 2;
    const float* h0src = (l == 0) ? h0in : h0mid;
    const float* h1src = (l == 0) ? h1in : h1mid;
    float* o0 = (l == 0) ? h0mid : (float*)d_out;
    float* o1 = (l == 0) ? h1mid : ((float*)d_out + (size_t)N * 32);
    const int applyAct = (l == 0) ? 1 : 0;

    // Hoisted node-level GEMMs on the WMMA pipe (K=32, fp32).
    // Fast fused path: shared A fragments across all 32-wide weights.
    GemmPtrs gp;
    gp.B[0] = Wq;   gp.D[0] = Qn;
    gp.B[1] = Wk00; gp.D[1] = K00n;
    gp.B[2] = Wv00; gp.D[2] = V00n;
    gp.B[3] = Ws0;  gp.D[3] = S0;
    gp.B[4] = Wv01; gp.D[4] = V01n;  // only used when cout1 == 32
    int nW = (cout1 == 32) ? 5 : 4;
    {
      int tiles = (N / 16) * 2;
      dim3 blk(32, 4), grd((tiles + 3) / 4);
      node_gemm_fused<<<grd, blk, 0, stream>>>(h0src, gp, nW, N);
    }
    if (cout1 != 32)  // layer 2: Wv01 is 32 x 2 -> masked generic path
      gemm(h0src, 32, 1, Wv01, cout1, 1, V01n, cout1, 1, N, 32, cout1);

    // einsum('nci,cd->ndi') as 3 strided GEMMs each for Wv11 and Ws1
    for (int i = 0; i < 3; ++i) {
      gemm(h1src + i, cin1 * 3, 3, Wv11, cout1, 1, V11n + i, cout1 * 3, 3, N,
           cin1, cout1);
      gemm(h1src + i, cin1 * 3, 3, Ws1, cout1, 1, S1 + i, cout1 * 3, 3, N,
           cin1, cout1);
    }

    init_mz_kernel<<<(N * 4 + 255) / 256, 256, 0, stream>>>(mbuf, zbuf, N * 4);

    if (l == 0) {
      edge_logits_kernel<3><<<egrd, eblk, 0, stream>>>(
          ef, dE, rhat, esrc, edst, h1src, Wr1, br1, Wr2, Wk10, Qn, K00n, gbuf,
          h1pbuf, attbuf, mbuf, E);
    } else {
      edge_logits_kernel<32><<<egrd, eblk, 0, stream>>>(
          ef, dE, rhat, esrc, edst, h1src, Wr1, br1, Wr2, Wk10, Qn, K00n, gbuf,
          h1pbuf, attbuf, mbuf, E);
    }

    edge_softmax_kernel<<<egrd, eblk, 0, stream>>>(edst, attbuf, mbuf, zbuf, E);

    if (l == 0) {
      edge_scatter_kernel<3, 32><<<egrd, eblk, 0, stream>>>(
          esrc, edst, rhat, gbuf, h1pbuf, attbuf, zbuf, V00n, V01n, V11n, Wv10,
          S0, S1, E);
    } else {
      edge_scatter_kernel<32, 2><<<egrd, eblk, 0, stream>>>(
          esrc, edst, rhat, gbuf, h1pbuf, attbuf, zbuf, V00n, V01n, V11n, Wv10,
          S0, S1, E);
    }

    node_finalize_kernel<<<ngrd, nblk, 0, stream>>>(S0, S1, cout1, o0, o1,
                                                    applyAct, N);
  }
}